// HGAD_31696858644805
// MI455X (gfx1250) — compile-verified
//
#include <hip/hip_runtime.h>

#define NN 100000
#define EE 800000
#define BN_EPS_ 1e-5f

typedef __attribute__((ext_vector_type(2))) float v2f;
typedef __attribute__((ext_vector_type(8))) float v8f;

// ---------------------------------------------------------------------------
// Y[N,C] = X[N,K] @ W[K,C] + bias   (fp32, exact via V_WMMA_F32_16X16X4_F32)
// One wave per 16-row strip: A fragments for the whole K are preloaded into
// registers once, then reused across all C/16 output tiles (8x less X traffic
// than tile-per-wave). W is <=64KB -> L2 resident.
//   A 16x4 : lane l (m=l&15, half=l>>4): v0 = X[m][k+2*half], v1 = X[m][k+2*half+1]
//   B 4x16 : lane l (n=l&15, half=l>>4): v0 = W[k+2*half][n], v1 = W[k+2*half+1][n]
//   D 16x16: VGPR r: row = r + 8*half, col = l&15
// ---------------------------------------------------------------------------
template <int K, int C>
__global__ void gemm_strip_wmma(const float* __restrict__ X,
                                const float* __restrict__ W,
                                const float* __restrict__ bias,
                                float* __restrict__ Y) {
  constexpr int NT = C / 16;
  constexpr int KS = K / 4;
  const int lane = threadIdx.x & 31;
  const int m16  = blockIdx.x << 4;
  const int half = lane >> 4;
  const int l    = lane & 15;

  const float* xrow = X + (size_t)(m16 + l) * K + (half << 1);
  v2f a[KS];
#pragma unroll
  for (int ks = 0; ks < KS; ++ks) {
    a[ks].x = xrow[4 * ks];
    a[ks].y = xrow[4 * ks + 1];
  }

#pragma unroll
  for (int t = 0; t < NT; ++t) {
    const int n16 = t << 4;
    v8f d = {0.f, 0.f, 0.f, 0.f, 0.f, 0.f, 0.f, 0.f};
#pragma unroll
    for (int ks = 0; ks < KS; ++ks) {
      const int ka = 4 * ks + (half << 1);
      v2f b;
      b.x = W[(size_t)ka * C + n16 + l];
      b.y = W[(size_t)(ka + 1) * C + n16 + l];
      d = __builtin_amdgcn_wmma_f32_16x16x4_f32(false, a[ks], false, b, (short)0,
                                                d, false, false);
    }
    const float bb = bias[n16 + l];
    float* yout = Y + (size_t)(m16 + (half << 3)) * C + n16 + l;
#pragma unroll
    for (int r = 0; r < 8; ++r) yout[(size_t)r * C] = d[r] + bb;
  }
}

// ---------------------------------------------------------------------------
// per-gatv2 init: acc = 0, lmax(enc) = 0 (== -inf in ordered encoding), den = 0
// ---------------------------------------------------------------------------
__global__ void gat_init_k(float* __restrict__ acc, unsigned* __restrict__ lmaxe,
                           float* __restrict__ den, int NC) {
  const int i = blockIdx.x * blockDim.x + threadIdx.x;
  if (i < NC) acc[i] = 0.f;
  if (i < NN) { lmaxe[i] = 0u; den[i] = 0.f; }
}

__device__ __forceinline__ unsigned f32_ord_enc(float x) {
  unsigned u = __float_as_uint(x);
  return (u & 0x80000000u) ? ~u : (u | 0x80000000u);
}
__device__ __forceinline__ float f32_ord_dec(unsigned u) {
  return __uint_as_float((u & 0x80000000u) ? (u & 0x7FFFFFFFu) : ~u);
}

// ---------------------------------------------------------------------------
// pass 1: one wave per edge; We (16xC <= 8KB) and att staged in LDS per block.
//   m = leaky_relu(XL[src] + XR[dst] + ea @ We, 0.2);  logit = m . att
//   segment-max over dst via ordered-int atomicMax
// ---------------------------------------------------------------------------
template <int C>
__global__ void edge_logit_k(const float* __restrict__ XL, const float* __restrict__ XR,
                             const int* __restrict__ esrc, const int* __restrict__ edst,
                             const float* __restrict__ EA, const float* __restrict__ We,
                             const float* __restrict__ att,
                             float* __restrict__ logit, unsigned* __restrict__ lmaxe) {
  __shared__ float sWe[16 * C];
  __shared__ float sAtt[C];
  for (int i = threadIdx.x; i < 16 * C; i += blockDim.x) sWe[i] = We[i];
  for (int i = threadIdx.x; i < C; i += blockDim.x) sAtt[i] = att[i];
  __syncthreads();

  const int e    = (int)((blockIdx.x * blockDim.x + threadIdx.x) >> 5);
  const int lane = threadIdx.x & 31;
  if (e >= EE) return;
  const int s = esrc[e], d = edst[e];
  constexpr int CPL = C / 32;

  const float ea_l = (lane < 16) ? EA[(size_t)e * 16 + lane] : 0.f;

  float part = 0.f;
#pragma unroll
  for (int j = 0; j < CPL; ++j) {
    const int c = lane + (j << 5);
    float v = XL[(size_t)s * C + c] + XR[(size_t)d * C + c];
    float ec = 0.f;
#pragma unroll
    for (int k = 0; k < 16; ++k) {
      const float eak = __shfl(ea_l, k, 32);
      ec = fmaf(eak, sWe[k * C + c], ec);
    }
    v += ec;
    v = v > 0.f ? v : 0.2f * v;
    part = fmaf(v, sAtt[c], part);
  }
#pragma unroll
  for (int off = 16; off; off >>= 1) part += __shfl_xor(part, off, 32);
  if (lane == 0) {
    logit[e] = part;
    atomicMax(lmaxe + d, f32_ord_enc(part));
  }
}

// pass 2: one thread per edge: exp(logit - lmax[dst]) and segment-sum denom
__global__ void edge_exp_k(const float* __restrict__ logit, const int* __restrict__ edst,
                           const unsigned* __restrict__ lmaxe,
                           float* __restrict__ expv, float* __restrict__ den) {
  const int e = blockIdx.x * blockDim.x + threadIdx.x;
  if (e >= EE) return;
  const int d = edst[e];
  const float lm = f32_ord_dec(lmaxe[d]);
  const float ex = __expf(logit[e] - lm);
  expv[e] = ex;
  atomicAdd(den + d, ex);
}

// pass 3: one wave per edge: acc[dst] += alpha * XL[src]
__global__ void edge_scatter_k(const float* __restrict__ XL,
                               const int* __restrict__ esrc, const int* __restrict__ edst,
                               const float* __restrict__ expv, const float* __restrict__ den,
                               int C, float* __restrict__ acc) {
  const int e    = (int)((blockIdx.x * blockDim.x + threadIdx.x) >> 5);
  const int lane = threadIdx.x & 31;
  if (e >= EE) return;
  const int s = esrc[e], d = edst[e];
  const float alpha = expv[e] / (den[d] + 1e-16f);
  const int cpl = C >> 5;
  for (int j = 0; j < cpl; ++j) {
    const int c = lane + (j << 5);
    atomicAdd(acc + (size_t)d * C + c, XL[(size_t)s * C + c] * alpha);
  }
}

// finalize: out = acc + b
__global__ void fin_plain_k(const float* __restrict__ acc, const float* __restrict__ b,
                            float* __restrict__ out, int C, int NC) {
  const int i = blockIdx.x * blockDim.x + threadIdx.x;
  if (i >= NC) return;
  out[i] = acc[i] + b[i % C];
}

// finalize: out = BN_eval(relu(acc + b))
__global__ void fin_bnrelu_k(const float* __restrict__ acc, const float* __restrict__ b,
                             const float* __restrict__ gamma, const float* __restrict__ beta,
                             const float* __restrict__ rm, const float* __restrict__ rv,
                             float* __restrict__ out, int C, int NC) {
  const int i = blockIdx.x * blockDim.x + threadIdx.x;
  if (i >= NC) return;
  const int c = i % C;
  float x = acc[i] + b[c];
  x = x > 0.f ? x : 0.f;
  out[i] = (x - rm[c]) * (gamma[c] * rsqrtf(rv[c] + BN_EPS_)) + beta[c];
}

// ---------------------------------------------------------------------------
// host side
// ---------------------------------------------------------------------------
struct GatP { const float *We, *Wl, *Wr, *att, *b, *bl, *br; };
struct BnP  { const float *beta, *gamma, *rm, *rv; };

struct Ws {
  float* XL; float* XR; float* ACC; float* BUF0; float* BUF1;
  float* LOGIT; float* EXPV; unsigned* LMAX; float* DEN;
};

static void launch_gemm(const float* X, const float* W, const float* b, float* Y,
                        int K, int C, hipStream_t st) {
  dim3 g(NN / 16), blk(32);
  if (K == 64 && C == 128)      gemm_strip_wmma<64, 128><<<g, blk, 0, st>>>(X, W, b, Y);
  else if (K == 32 && C == 128) gemm_strip_wmma<32, 128><<<g, blk, 0, st>>>(X, W, b, Y);
  else if (K == 128 && C == 64) gemm_strip_wmma<128, 64><<<g, blk, 0, st>>>(X, W, b, Y);
  else if (K == 128 && C == 32) gemm_strip_wmma<128, 32><<<g, blk, 0, st>>>(X, W, b, Y);
}

static void launch_logit(const Ws& w, const int* es, const int* ed, const float* ea,
                         const GatP& g, int C, hipStream_t st) {
  dim3 gr(EE / 8), blk(256);
  if (C == 128)     edge_logit_k<128><<<gr, blk, 0, st>>>(w.XL, w.XR, es, ed, ea, g.We, g.att, w.LOGIT, w.LMAX);
  else if (C == 64) edge_logit_k<64><<<gr, blk, 0, st>>>(w.XL, w.XR, es, ed, ea, g.We, g.att, w.LOGIT, w.LMAX);
  else              edge_logit_k<32><<<gr, blk, 0, st>>>(w.XL, w.XR, es, ed, ea, g.We, g.att, w.LOGIT, w.LMAX);
}

static void run_gat(const float* xs, int dsrc, const float* xd, int ddst, int C,
                    const int* es, const int* ed, const float* ea, const GatP& g,
                    const Ws& w, hipStream_t st) {
  launch_gemm(xs, g.Wl, g.bl, w.XL, dsrc, C, st);
  launch_gemm(xd, g.Wr, g.br, w.XR, ddst, C, st);
  const int NC = NN * C;
  gat_init_k<<<(NC + 255) / 256, 256, 0, st>>>(w.ACC, w.LMAX, w.DEN, NC);
  launch_logit(w, es, ed, ea, g, C, st);
  edge_exp_k<<<(EE + 255) / 256, 256, 0, st>>>(w.LOGIT, ed, w.LMAX, w.EXPV, w.DEN);
  edge_scatter_k<<<EE / 8, 256, 0, st>>>(w.XL, es, ed, w.EXPV, w.DEN, C, w.ACC);
}

extern "C" void kernel_launch(void* const* d_in, const int* in_sizes, int n_in,
                              void* d_out, int out_size, void* d_ws, size_t ws_size,
                              hipStream_t stream) {
  // -------- classify inputs by flat element count (order-robust) --------
  const float *x_claim = nullptr, *x_party = nullptr, *ea_cp = nullptr, *ea_pc = nullptr;
  const int *ei_cp = nullptr, *ei_pc = nullptr;
  const float* pl[80]; int plsz[80]; int np = 0;
  for (int i = 0; i < n_in; ++i) {
    const int sz = in_sizes[i];
    const void* p = d_in[i];
    if (sz == NN * 64)       x_claim = (const float*)p;
    else if (sz == NN * 32)  x_party = (const float*)p;
    else if (sz == 2 * EE)   { if (!ei_cp) ei_cp = (const int*)p; else ei_pc = (const int*)p; }
    else if (sz == EE * 16)  { if (!ea_cp) ea_cp = (const float*)p; else ea_pc = (const float*)p; }
    else if (np < 80)        { plsz[np] = sz; pl[np++] = (const float*)p; }
  }
  // sorted-key pytree order -> first param leaf is bn0_claim.beta (128 elems);
  // insertion order -> first leaf is enc0_cp.Wl (8192 elems)
  const bool sorted = (np > 0 && plsz[0] == 128);

  auto G = [&](int base) -> GatP {
    GatP g;
    if (sorted) { g.We = pl[base]; g.Wl = pl[base+1]; g.Wr = pl[base+2];
                  g.att = pl[base+3]; g.b = pl[base+4]; g.bl = pl[base+5]; g.br = pl[base+6]; }
    else        { g.Wl = pl[base]; g.bl = pl[base+1]; g.Wr = pl[base+2]; g.br = pl[base+3];
                  g.We = pl[base+4]; g.att = pl[base+5]; g.b = pl[base+6]; }
    return g;
  };
  auto B = [&](int base) -> BnP {
    if (sorted) return BnP{pl[base], pl[base+1], pl[base+2], pl[base+3]};   // beta,gamma,rm,rv
    return BnP{pl[base+1], pl[base], pl[base+2], pl[base+3]};               // gamma,beta,rm,rv
  };

  GatP enc0_cp, enc0_pc, enc1_cp, enc1_pc, dec0_cp, dec0_pc, dec1_cp, dec1_pc;
  BnP bn0_claim, bn0_party, bnd_claim, bnd_party;
  if (sorted) {
    bn0_claim = B(0);  bn0_party = B(4);  bnd_claim = B(8);  bnd_party = B(12);
    dec0_cp = G(16); dec0_pc = G(23); dec1_cp = G(30); dec1_pc = G(37);
    enc0_cp = G(44); enc0_pc = G(51); enc1_cp = G(58); enc1_pc = G(65);
  } else {
    enc0_cp = G(0);  enc0_pc = G(7);  bn0_claim = B(14); bn0_party = B(18);
    enc1_cp = G(22); enc1_pc = G(29); dec0_cp = G(36);  dec0_pc = G(43);
    bnd_claim = B(50); bnd_party = B(54); dec1_cp = G(58); dec1_pc = G(65);
  }

  const int* cp_src = ei_cp;        const int* cp_dst = ei_cp + EE;
  const int* pc_src = ei_pc;        const int* pc_dst = ei_pc + EE;

  // -------- workspace layout (floats) --------
  float* W0 = (float*)d_ws;
  Ws w;
  w.XL    = W0;
  w.XR    = W0 + 12800000;
  w.ACC   = W0 + 25600000;
  w.BUF0  = W0 + 38400000;    // claim-side hidden (N x 128)
  w.BUF1  = W0 + 51200000;    // party-side hidden (N x 128)
  w.LOGIT = W0 + 64000000;
  w.EXPV  = W0 + 64800000;
  w.LMAX  = (unsigned*)(W0 + 65600000);
  w.DEN   = W0 + 65700000;

  float* out       = (float*)d_out;
  float* z_claim   = out;                  // N x 64
  float* z_party   = out + (size_t)NN * 64;
  float* rec_claim = out + (size_t)NN * 128;
  float* rec_party = out + (size_t)NN * 192;

  const int T = 256;
  auto NCg = [&](int C) { return (NN * C + T - 1) / T; };

  // ================= encoder layer 0 (out 128, relu+BN) =================
  run_gat(x_claim, 64, x_party, 32, 128, cp_src, cp_dst, ea_cp, enc0_cp, w, stream);
  fin_bnrelu_k<<<NCg(128), T, 0, stream>>>(w.ACC, enc0_cp.b, bn0_party.gamma, bn0_party.beta,
                                           bn0_party.rm, bn0_party.rv, w.BUF1, 128, NN * 128);
  run_gat(x_party, 32, x_claim, 64, 128, pc_src, pc_dst, ea_pc, enc0_pc, w, stream);
  fin_bnrelu_k<<<NCg(128), T, 0, stream>>>(w.ACC, enc0_pc.b, bn0_claim.gamma, bn0_claim.beta,
                                           bn0_claim.rm, bn0_claim.rv, w.BUF0, 128, NN * 128);

  // ================= encoder layer 1 (latent, out 64) =================
  run_gat(w.BUF0, 128, w.BUF1, 128, 64, cp_src, cp_dst, ea_cp, enc1_cp, w, stream);
  fin_plain_k<<<NCg(64), T, 0, stream>>>(w.ACC, enc1_cp.b, z_party, 64, NN * 64);
  run_gat(w.BUF1, 128, w.BUF0, 128, 64, pc_src, pc_dst, ea_pc, enc1_pc, w, stream);
  fin_plain_k<<<NCg(64), T, 0, stream>>>(w.ACC, enc1_pc.b, z_claim, 64, NN * 64);

  // ================= decoder hidden (out 128, relu+BN) =================
  run_gat(z_claim, 64, z_party, 64, 128, cp_src, cp_dst, ea_cp, dec0_cp, w, stream);
  fin_bnrelu_k<<<NCg(128), T, 0, stream>>>(w.ACC, dec0_cp.b, bnd_party.gamma, bnd_party.beta,
                                           bnd_party.rm, bnd_party.rv, w.BUF1, 128, NN * 128);
  run_gat(z_party, 64, z_claim, 64, 128, pc_src, pc_dst, ea_pc, dec0_pc, w, stream);
  fin_bnrelu_k<<<NCg(128), T, 0, stream>>>(w.ACC, dec0_pc.b, bnd_claim.gamma, bnd_claim.beta,
                                           bnd_claim.rm, bnd_claim.rv, w.BUF0, 128, NN * 128);

  // ================= decoder output =================
  run_gat(w.BUF0, 128, w.BUF1, 128, 32, cp_src, cp_dst, ea_cp, dec1_cp, w, stream);
  fin_plain_k<<<NCg(32), T, 0, stream>>>(w.ACC, dec1_cp.b, rec_party, 32, NN * 32);
  run_gat(w.BUF1, 128, w.BUF0, 128, 64, pc_src, pc_dst, ea_pc, dec1_pc, w, stream);
  fin_plain_k<<<NCg(64), T, 0, stream>>>(w.ACC, dec1_pc.b, rec_claim, 64, NN * 64);
}